// ReconstructionLoss_34256659152993
// MI455X (gfx1250) — compile-verified
//
#include <hip/hip_runtime.h>

// ---------------------------------------------------------------------------
// ReconstructionLoss for MI455X (gfx1250, wave32)
//   pass 1: one wave32 per 8x8x8 block -> per-block mean |gt-pred|.
//           global_load_b128 gathers; ds_swizzle_b32 XOR-butterfly reduce
//           (1 DS + 1 VALU per step, full-EXEC waves).
//   pass 2: single-workgroup deterministic reduce; per-wave 32-lane fold via
//           V_WMMA_F32_16X16X4_F32 (A = ones -> column-sum of B), then a
//           swizzle butterfly.  No float atomics -> bitwise-deterministic.
// Roofline: ~132 MB gathered reads / 23.3 TB/s HBM (and mostly L2-resident:
// indices only span 4096 distinct blocks ~= 16 MB unique) -> pure bandwidth
// problem; compute path kept minimal.
// ---------------------------------------------------------------------------

typedef __attribute__((ext_vector_type(2))) float v2f;
typedef __attribute__((ext_vector_type(8))) float v8f;

#define DIM_D 160
#define DIM_H 160
#define DIM_W 160
#define BS 8
#define WAVES_PER_WG 8   // 256 threads = 8 wave32s

// ds_swizzle_b32 group-of-32 XOR butterfly: lane' = lane ^ MASK
// offset encoding: [15]=0, xor_mask=[14:10], or_mask=[9:5]=0, and_mask=[4:0]=0x1f
#define SWZ_XOR(v, MASK) \
    __int_as_float(__builtin_amdgcn_ds_swizzle(__float_as_int(v), (((MASK) << 10) | 0x1f)))

__device__ __forceinline__ float wave32_sum(float v) {
    v += SWZ_XOR(v, 16);
    v += SWZ_XOR(v, 8);
    v += SWZ_XOR(v, 4);
    v += SWZ_XOR(v, 2);
    v += SWZ_XOR(v, 1);
    return v;   // full sum in every lane
}

__global__ __launch_bounds__(256) void block_l1_kernel(
    const float* __restrict__ masked,
    const float* __restrict__ orig,
    const int*   __restrict__ mi,      // [N,5]: sample, _, bz, by, bx
    float*       __restrict__ block_means,
    int n_blocks)
{
    const int wave = threadIdx.x >> 5;
    const int lane = threadIdx.x & 31;
    const int blk  = blockIdx.x * WAVES_PER_WG + wave;
    if (blk >= n_blocks) return;   // wave-uniform guard: active waves keep full EXEC

    const int* id = mi + (size_t)blk * 5;
    const int b  = id[0];
    const int bz = id[2];
    const int by = id[3];
    const int bx = id[4];

    const size_t HW   = (size_t)DIM_H * DIM_W;
    const size_t base = (size_t)b * DIM_D * HW
                      + (size_t)(bz * BS) * HW
                      + (size_t)(by * BS) * DIM_W
                      + (size_t)(bx * BS);

    // 64 rows of 8 contiguous floats per block; each lane handles 2 rows.
    float acc = 0.0f;
#pragma unroll
    for (int r = 0; r < 2; ++r) {
        const int row = (lane << 1) + r;           // 0..63
        const int dz  = row >> 3;
        const int dy  = row & 7;
        const size_t off = base + (size_t)dz * HW + (size_t)dy * DIM_W;

        const float4* mp = (const float4*)(masked + off);  // 16B aligned (x = bx*8)
        const float4* op = (const float4*)(orig   + off);
        float4 m0 = mp[0], m1 = mp[1];
        float4 o0 = op[0], o1 = op[1];
        acc += fabsf(o0.x - m0.x) + fabsf(o0.y - m0.y) +
               fabsf(o0.z - m0.z) + fabsf(o0.w - m0.w);
        acc += fabsf(o1.x - m1.x) + fabsf(o1.y - m1.y) +
               fabsf(o1.z - m1.z) + fabsf(o1.w - m1.w);
    }

    const float block_sum = wave32_sum(acc);       // deterministic tree
    if (lane == 0)
        block_means[blk] = block_sum * (1.0f / (BS * BS * BS));
}

__global__ __launch_bounds__(1024) void reduce_kernel(
    const float* __restrict__ block_means,
    int n,
    float* __restrict__ out)
{
    const int tid  = threadIdx.x;
    const int wave = tid >> 5;
    const int lane = tid & 31;

    float acc = 0.0f;
    for (int i = tid; i < n; i += 1024)
        acc += block_means[i];

    // --- WMMA lane reduction: D = A(ones 16x4) x B + 0  => D[m][n] = sum_k B[k][n].
    // Each lane deposits its partial into one distinct B slot (b.x), b.y = 0.
    // A all-ones makes the result invariant to B's lane->(k,n) mapping: summing
    // D's VGPR0 across all 32 lanes counts every B slot once per D-row, and
    // VGPR0 holds rows M=0 and M=8 (identical) -> lane-sum == 2 * waveSum.
    // Executed uniformly by every wave (EXEC all ones; no divergence here).
    v2f a;  a.x = 1.0f; a.y = 1.0f;       // A = ones(16x4)
    v2f bv; bv.x = acc; bv.y = 0.0f;      // B holds the 32 partials (+32 zeros)
    v8f c = {};
    v8f d = __builtin_amdgcn_wmma_f32_16x16x4_f32(
        /*neg_a=*/false, a, /*neg_b=*/false, bv,
        /*c_mod=*/(short)0, c, /*reuse_a=*/false, /*reuse_b=*/false);

    const float wave_sum = wave32_sum(d[0]) * 0.5f;  // undo row-0/row-8 duplication

    __shared__ float lds[32];
    if (lane == 0) lds[wave] = wave_sum;
    __syncthreads();

    if (tid == 0) {
        float total = 0.0f;
#pragma unroll
        for (int i = 0; i < 32; ++i) total += lds[i];
        out[0] = total / (float)n;
    }
}

extern "C" void kernel_launch(void* const* d_in, const int* in_sizes, int n_in,
                              void* d_out, int out_size, void* d_ws, size_t ws_size,
                              hipStream_t stream) {
    const float* masked = (const float*)d_in[0];
    const float* orig   = (const float*)d_in[1];
    const int*   mi     = (const int*)d_in[2];   // integer input -> const int*
    const int n_blocks  = in_sizes[2] / 5;       // 32000

    float* block_means = (float*)d_ws;           // n_blocks floats (128 KB)

    const int grid = (n_blocks + WAVES_PER_WG - 1) / WAVES_PER_WG;
    block_l1_kernel<<<grid, 256, 0, stream>>>(masked, orig, mi, block_means, n_blocks);
    reduce_kernel<<<1, 1024, 0, stream>>>(block_means, n_blocks, (float*)d_out);
}